// SinLayer_35124242547412
// MI455X (gfx1250) — compile-verified
//
#include <hip/hip_runtime.h>
#include <math.h>
#include <stdint.h>

// Problem constants (match reference setup_inputs)
#define LUT_RES   512
#define TWO_PI_F  6.28318530717958647692f
#define IDX_SCALE ((float)LUT_RES / TWO_PI_F)   // 512 / 2pi

#define E_DIM 128
#define N_DIM 512
#define S_DIM 1024

// Tiling
#define RT      16      // rows (flattened b*s) per block
#define NT      128     // n columns per block
#define THREADS 256
#define ROWS_PER_THREAD 8   // RT * NT / THREADS

__global__ __launch_bounds__(THREADS)
void sin_lut_layer_kernel(const float* __restrict__ x,   // (R, E) rows contiguous
                          const float* __restrict__ w,   // (E, N)
                          const float* __restrict__ b,   // (E, N)
                          const float* __restrict__ t,   // (S)
                          float* __restrict__ out)       // (R, N)
{
    __shared__ __align__(16) float lut[LUT_RES];
    __shared__ __align__(16) float xs[RT * E_DIM];   // 8 KB x tile
    __shared__ float ts[RT];                          // t * IDX_SCALE per row

    const int tid      = threadIdx.x;
    const int row_base = blockIdx.y * RT;
    const int n0       = blockIdx.x * NT + (tid & (NT - 1));
    const int rsub     = tid >> 7;                    // 0 or 1

    // ---- Stage x tile into LDS with the CDNA5 async copy engine ----------
    // Tile is contiguous in global memory: RT*E floats = 8192 bytes.
    // 256 threads * 16B * 2 ops covers it.  INST_OFFSET applies to both the
    // LDS and global address per the ISA pseudocode.
    {
        uint32_t lds_addr = (uint32_t)(uintptr_t)(&xs[0]) + (uint32_t)tid * 16u;
        uint64_t gaddr    = (uint64_t)(uintptr_t)(x + (size_t)row_base * E_DIM)
                            + (uint64_t)tid * 16u;
        asm volatile("global_load_async_to_lds_b128 %0, %1, off"
                     :: "v"(lds_addr), "v"(gaddr) : "memory");
        asm volatile("global_load_async_to_lds_b128 %0, %1, off offset:4096"
                     :: "v"(lds_addr), "v"(gaddr) : "memory");
    }

    // ---- Build the sin LUT in LDS (v_sin_f32) -----------------------------
    for (int i = tid; i < LUT_RES; i += THREADS)
        lut[i] = __sinf((float)i * (TWO_PI_F / (float)LUT_RES));

    // ---- Pre-scaled t per row --------------------------------------------
    if (tid < RT) {
        int row = row_base + tid;
        ts[tid] = t[row & (S_DIM - 1)] * IDX_SCALE;   // s = row % S
    }

    asm volatile("s_wait_asynccnt 0" ::: "memory");
    __syncthreads();

    float acc[ROWS_PER_THREAD];
    float tv[ROWS_PER_THREAD];
#pragma unroll
    for (int j = 0; j < ROWS_PER_THREAD; ++j) {
        acc[j] = 0.0f;
        tv[j]  = ts[rsub + 2 * j];
    }

    const float* __restrict__ wp = w + n0;   // column n0, stride N
    const float* __restrict__ bp = b + n0;

    for (int e = 0; e < E_DIM; e += 4) {
        if (e + 8 < E_DIM) {
            __builtin_prefetch(&wp[(size_t)(e + 8) * N_DIM], 0, 0);
            __builtin_prefetch(&bp[(size_t)(e + 8) * N_DIM], 0, 0);
        }
        float we[4], be[4];
#pragma unroll
        for (int q = 0; q < 4; ++q) {
            we[q] = wp[(size_t)(e + q) * N_DIM] * IDX_SCALE;  // L2-resident, coalesced
            be[q] = bp[(size_t)(e + q) * N_DIM] * IDX_SCALE;
        }
#pragma unroll
        for (int j = 0; j < ROWS_PER_THREAD; ++j) {
            const int r = rsub + 2 * j;
            float4 xv = *(const float4*)&xs[r * E_DIM + e];   // ds_load_b128 (wave-uniform)
            float bt0 = be[0] + tv[j];
            float bt1 = be[1] + tv[j];
            float bt2 = be[2] + tv[j];
            float bt3 = be[3] + tv[j];
            // idx = floor(phase * 512/2pi) mod 512  ==  floor(frac(phase/2pi)*512) mod 512
            int i0 = __float2int_rd(fmaf(we[0], xv.x, bt0)) & (LUT_RES - 1);
            int i1 = __float2int_rd(fmaf(we[1], xv.y, bt1)) & (LUT_RES - 1);
            int i2 = __float2int_rd(fmaf(we[2], xv.z, bt2)) & (LUT_RES - 1);
            int i3 = __float2int_rd(fmaf(we[3], xv.w, bt3)) & (LUT_RES - 1);
            acc[j] += lut[i0];
            acc[j] += lut[i1];
            acc[j] += lut[i2];
            acc[j] += lut[i3];
        }
    }

#pragma unroll
    for (int j = 0; j < ROWS_PER_THREAD; ++j) {
        const int r = rsub + 2 * j;
        out[(size_t)(row_base + r) * N_DIM + n0] = acc[j];
    }
}

extern "C" void kernel_launch(void* const* d_in, const int* in_sizes, int n_in,
                              void* d_out, int out_size, void* d_ws, size_t ws_size,
                              hipStream_t stream) {
    (void)in_sizes; (void)n_in; (void)d_ws; (void)ws_size;
    const float* x = (const float*)d_in[0];   // (B,S,E)
    const float* w = (const float*)d_in[1];   // (E,N)
    const float* b = (const float*)d_in[2];   // (E,N)
    const float* t = (const float*)d_in[3];   // (S)
    float* out = (float*)d_out;               // (B,S,N)

    const int R = out_size / N_DIM;           // B*S = 2048
    dim3 grid(N_DIM / NT, R / RT);            // (4, 128)
    dim3 block(THREADS);
    sin_lut_layer_kernel<<<grid, block, 0, stream>>>(x, w, b, t, out);
}